// DChord_46033459478542
// MI455X (gfx1250) — compile-verified
//
#include <hip/hip_runtime.h>

typedef float v2f __attribute__((ext_vector_type(2)));
typedef float v8f __attribute__((ext_vector_type(8)));

#define CHORD_EPS 1e-4f

__device__ __forceinline__ float fbits(int i) { return __builtin_bit_cast(float, i); }
__device__ __forceinline__ int   ibits(float f) { return __builtin_bit_cast(int, f); }

// One-instruction abs-max: v_max_num_f32 with |.| source modifiers.
__device__ __forceinline__ float absmax(float a, float b) {
  float r;
  asm("v_max_num_f32_e64 %0, |%1|, |%2|" : "=v"(r) : "v"(a), "v"(b));
  return r;
}

// One-instruction DPP lane-max: v_max_num_f32 with DPP16 on src0.
#define MAX_DPP(x, CTRL)                                                     \
  ({                                                                         \
    float _r;                                                                \
    asm("v_max_num_f32_dpp %0, %1, %2 " CTRL                                 \
        " row_mask:0xf bank_mask:0xf bound_ctrl:1"                           \
        : "=v"(_r) : "v"(x), "v"(x));                                        \
    _r;                                                                      \
  })

// Swap the two 16-lane halves of the wave (ds_swizzle SWAPX16: xor_mask=0x10).
__device__ __forceinline__ float swapx16(float x) {
  return fbits(__builtin_amdgcn_ds_swizzle(ibits(x), 0x401F));
}

__global__ __launch_bounds__(256) void chord_wmma_kernel(
    const float* __restrict__ x, const float* __restrict__ tmpl,
    float* __restrict__ out, int nrows) {
  const int lane  = threadIdx.x & 31;
  const int n0    = lane & 15;   // N-tile column / A-row handled by this lane
  const int khalf = lane >> 4;   // K-half of each 4-wide K chunk
  const int wavesPerBlock = blockDim.x >> 5;
  const int waveId = blockIdx.x * wavesPerBlock + (threadIdx.x >> 5);
  const int nWaves = gridDim.x * wavesPerBlock;
  const int nTiles = (nrows + 15) >> 4;

  // ---- B operand (templates^T) in registers, V_WMMA_F32_16X16X4_F32 layout ----
  // chunk c covers K=4c..4c+3; VGPR j of lane (n0,khalf) holds T[n][4c+2*khalf+j].
  // Columns 25..31 padded with 0 -> d1 columns 25..31 are exactly 0 after WMMA.
  float b0[3][2], b1[3][2];
#pragma unroll
  for (int c = 0; c < 3; ++c) {
#pragma unroll
    for (int j = 0; j < 2; ++j) {
      const int k = 4 * c + 2 * khalf + j;
      b0[c][j] = tmpl[n0 * 12 + k];
      b1[c][j] = (n0 + 16 < 25) ? tmpl[(n0 + 16) * 12 + k] : 0.0f;
    }
  }

  const float unit = 0.2886751345948129f;  // 1/sqrt(12)
  const bool  hasHiCol = (n0 < 9);         // lane owns a valid column in N-tile 1

  for (int tile = waveId; tile < nTiles; tile += nWaves) {
    const int row0 = tile << 4;
    const bool full = (row0 + 16 <= nrows);  // uniform per wave

    // ---- load 6 of the row's 12 values (lane pair L / L^16 covers all 12) ----
    float a[3][2];
    if (full) {
      const float* xr = x + (size_t)(row0 + n0) * 12 + 2 * khalf;
#pragma unroll
      for (int c = 0; c < 3; ++c) {
        v2f v = __builtin_nontemporal_load((const v2f*)(xr + 4 * c));
        a[c][0] = v.x; a[c][1] = v.y;
      }
    } else {
#pragma unroll
      for (int c = 0; c < 3; ++c) { a[c][0] = 0.0f; a[c][1] = 0.0f; }
      if (row0 + n0 < nrows) {
        const float* xr = x + (size_t)(row0 + n0) * 12 + 2 * khalf;
#pragma unroll
        for (int c = 0; c < 3; ++c) {
          v2f v = __builtin_nontemporal_load((const v2f*)(xr + 4 * c));
          a[c][0] = v.x; a[c][1] = v.y;
        }
      }
    }

    // ---- thresholded L2 normalize (norm>EPS  <=>  ss>EPS^2) ----
    float ss = 0.0f;
#pragma unroll
    for (int c = 0; c < 3; ++c) {
      ss = fmaf(a[c][0], a[c][0], ss);
      ss = fmaf(a[c][1], a[c][1], ss);
    }
    ss += swapx16(ss);  // combine the two K-halves of the row
    const bool  low   = !(ss > CHORD_EPS * CHORD_EPS);
    const float scale = __builtin_amdgcn_rsqf(ss);  // 1/norm, 1-ulp
    const float smul  = low ? 0.0f : scale;
    const float sadd  = low ? unit : 0.0f;
#pragma unroll
    for (int c = 0; c < 3; ++c)
#pragma unroll
      for (int j = 0; j < 2; ++j)
        a[c][j] = fmaf(a[c][j], smul, sadd);

    // ---- sim = xn @ templates^T : 3 chained K=4 fp32 WMMAs x 2 N-tiles ----
    v8f d0 = {}; v8f d1 = {};
#pragma unroll
    for (int c = 0; c < 3; ++c) {
      v2f av  = { a[c][0],  a[c][1]  };
      v2f bv0 = { b0[c][0], b0[c][1] };
      v2f bv1 = { b1[c][0], b1[c][1] };
      d0 = __builtin_amdgcn_wmma_f32_16x16x4_f32(false, av, false, bv0,
                                                 (short)0, d0, false, false);
      d1 = __builtin_amdgcn_wmma_f32_16x16x4_f32(false, av, false, bv1,
                                                 (short)0, d1, false, false);
    }

    // ---- thresholded inf-normalize over 25 templates (fused DPP butterfly) ----
    float o0v[8], o1v[8];
#pragma unroll
    for (int r = 0; r < 8; ++r) {
      const float s0 = d0[r];
      const float s1 = d1[r];              // cols 25..31 are exact 0 (padded B)
      float am = absmax(s0, s1);           // v_max_num_f32 |s0|,|s1|
      am = MAX_DPP(am, "quad_perm:[1,0,3,2]");
      am = MAX_DPP(am, "quad_perm:[2,3,0,1]");
      am = MAX_DPP(am, "row_ror:4");
      am = MAX_DPP(am, "row_ror:8");       // full 16-lane (25-col) row max
      const float rim = __builtin_amdgcn_rcpf(fmaxf(am, CHORD_EPS));
      const bool  lo2 = !(am > CHORD_EPS);
      o0v[r] = lo2 ? 1.0f : s0 * rim;
      o1v[r] = lo2 ? 1.0f : s1 * rim;
    }

    // ---- stores: one base pointer, immediate offsets; hoisted guards ----
    float* ob = out + (size_t)(row0 + 8 * khalf) * 25 + n0;
    if (full) {
#pragma unroll
      for (int r = 0; r < 8; ++r)
        __builtin_nontemporal_store(o0v[r], ob + r * 25);
      if (hasHiCol) {
#pragma unroll
        for (int r = 0; r < 8; ++r)
          __builtin_nontemporal_store(o1v[r], ob + 16 + r * 25);
      }
    } else {
#pragma unroll
      for (int r = 0; r < 8; ++r) {
        if (row0 + 8 * khalf + r < nrows) {
          __builtin_nontemporal_store(o0v[r], ob + r * 25);
          if (hasHiCol) __builtin_nontemporal_store(o1v[r], ob + 16 + r * 25);
        }
      }
    }
  }
}

extern "C" void kernel_launch(void* const* d_in, const int* in_sizes, int n_in,
                              void* d_out, int out_size, void* d_ws, size_t ws_size,
                              hipStream_t stream) {
  const float* x    = (const float*)d_in[0];   // (16,1,200000,12) fp32
  const float* tmpl = (const float*)d_in[1];   // (25,12) fp32
  float* out        = (float*)d_out;           // (16,1,200000,25) fp32

  const int nrows = in_sizes[0] / 12;          // B*T rows
  const int nTiles = (nrows + 15) / 16;
  const int threads = 256;                     // 8 waves per block (wave32)
  const int wavesPerBlock = threads / 32;
  const int tilesPerWave = 4;                  // amortize template prologue
  int blocks = (nTiles + wavesPerBlock * tilesPerWave - 1) /
               (wavesPerBlock * tilesPerWave);
  if (blocks < 1) blocks = 1;

  chord_wmma_kernel<<<blocks, threads, 0, stream>>>(x, tmpl, out, nrows);
}